// Encoder_9096740733413
// MI455X (gfx1250) — compile-verified
//
#include <hip/hip_runtime.h>

typedef float v2f __attribute__((ext_vector_type(2)));
typedef float v8f __attribute__((ext_vector_type(8)));

// ---------------------------------------------------------------- degree ----
__global__ void k_deg_init(float* __restrict__ deg, int N) {
    int i = blockIdx.x * blockDim.x + threadIdx.x;
    if (i < N) deg[i] = 1.0f;  // self-loop contributes 1
}

__global__ void k_deg_count(const long long* __restrict__ dst,
                            float* __restrict__ deg, int E) {
    int i = blockIdx.x * blockDim.x + threadIdx.x;
    if (i < E) atomicAdd(&deg[(int)dst[i]], 1.0f);
}

__global__ void k_dinv(const float* __restrict__ deg,
                       float* __restrict__ dinv, int N) {
    int i = blockIdx.x * blockDim.x + threadIdx.x;
    if (i < N) dinv[i] = rsqrtf(deg[i]);  // deg >= 1 always
}

// ------------------------------------------------------- WMMA fp32 GEMM ----
// C[M,Ncols] = A[M,K] @ B[K,Ncols], fp32 via V_WMMA_F32_16X16X4_F32.
// One 16x16 output tile per wave32. M, Ncols, K all multiples of 16/4 here.
__global__ void k_gemm_wmma_f32(const float* __restrict__ A,
                                const float* __restrict__ B,
                                float* __restrict__ C,
                                int M, int Ncols, int K) {
    const int lane = threadIdx.x & 31;
    const int wave = (blockIdx.x * blockDim.x + threadIdx.x) >> 5;
    const int ntn  = Ncols >> 4;
    const int ntm  = M >> 4;
    if (wave >= ntm * ntn) return;           // whole-wave uniform exit

    const int tm = wave / ntn;
    const int tn = wave % ntn;
    const int m0 = tm << 4;
    const int n0 = tn << 4;

    const int hl  = lane >> 4;               // half-wave select (K pair)
    const int l15 = lane & 15;

    // A fragment (16x4 f32): lanes 0-15 hold M=lane, VGPR0/1 = K=0/1;
    // lanes 16-31 hold M=lane-16, VGPR0/1 = K=2/3.
    const float* arow = A + (size_t)(m0 + l15) * K + 2 * hl;
    // B fragment (4x16 f32), mirrored: lane = N, half-wave selects K pair.
    const float* bcol = B + (size_t)(2 * hl) * Ncols + n0 + l15;

    v8f acc = {};
    #pragma unroll 4
    for (int k = 0; k < K; k += 4) {
        v2f a, b;
        a.x = arow[k];
        a.y = arow[k + 1];
        b.x = bcol[(size_t)k * Ncols];
        b.y = bcol[(size_t)(k + 1) * Ncols];
        acc = __builtin_amdgcn_wmma_f32_16x16x4_f32(
            /*neg_a=*/false, a, /*neg_b=*/false, b,
            /*c_mod=*/(short)0, acc, /*reuse_a=*/false, /*reuse_b=*/false);
    }

    // C/D layout: VGPR r, lanes 0-15 -> row m0+r; lanes 16-31 -> row m0+8+r.
    float* crow = C + (size_t)(m0 + 8 * hl) * Ncols + n0 + l15;
    #pragma unroll
    for (int r = 0; r < 8; ++r)
        crow[(size_t)r * Ncols] = acc[r];
}

// --------------------------------------------- aggregation (scatter-add) ----
// agg[n,f] = dinv[n]^2 * h[n,f] + bias[f]   (self-loop + bias folded in)
__global__ void k_init_agg(const float* __restrict__ h,
                           const float* __restrict__ dinv,
                           const float* __restrict__ bias,
                           float* __restrict__ agg, int N, int F) {
    int i = blockIdx.x * blockDim.x + threadIdx.x;
    if (i >= N * F) return;
    int n = i / F;
    int f = i - n * F;
    float di = dinv[n];
    agg[i] = di * di * h[i] + bias[f];
}

// agg[dst] += dinv[src]*dinv[dst] * h[src]; one wave per edge, float4 gather,
// scalar f32 atomics (global_atomic_add_f32, resolved in L2).
__global__ void k_edge_scatter(const long long* __restrict__ src,
                               const long long* __restrict__ dst,
                               const float* __restrict__ h,
                               const float* __restrict__ dinv,
                               float* __restrict__ agg, int E, int F) {
    int t    = blockIdx.x * blockDim.x + threadIdx.x;
    int e    = t >> 5;
    int lane = t & 31;
    if (e >= E) return;
    int s = (int)src[e];
    int d = (int)dst[e];
    float c = dinv[s] * dinv[d];
    const float4* hs = (const float4*)(h + (size_t)s * F);
    float* ad = agg + (size_t)d * F;
    for (int q = lane; q * 4 < F; q += 32) {
        float4 v = hs[q];
        int b = q * 4;
        atomicAdd(ad + b + 0, c * v.x);
        atomicAdd(ad + b + 1, c * v.y);
        atomicAdd(ad + b + 2, c * v.z);
        atomicAdd(ad + b + 3, c * v.w);
    }
}

__global__ void k_relu(const float* __restrict__ in, float* __restrict__ out,
                       int n) {
    int i = blockIdx.x * blockDim.x + threadIdx.x;
    if (i < n) out[i] = fmaxf(in[i], 0.0f);
}

// -------------------------------------------------------------- launcher ----
extern "C" void kernel_launch(void* const* d_in, const int* in_sizes, int n_in,
                              void* d_out, int out_size, void* d_ws, size_t ws_size,
                              hipStream_t stream) {
    const float*     x   = (const float*)d_in[0];
    const long long* ei  = (const long long*)d_in[1];  // int64 edge_index [2,E]
    const float*     W1  = (const float*)d_in[2];
    const float*     b1  = (const float*)d_in[3];
    const float*     W2  = (const float*)d_in[4];
    const float*     b2  = (const float*)d_in[5];
    float*           out = (float*)d_out;

    const int HID = in_sizes[3];            // 256
    const int OUT = in_sizes[5];            // 128
    const int IN  = in_sizes[2] / HID;      // 256
    const int N   = in_sizes[0] / IN;       // 100000
    const int E   = in_sizes[1] / 2;        // 3200000

    const long long* srcp = ei;
    const long long* dstp = ei + E;

    // workspace carve-out (256B aligned): deg, dinv, h[N*HID], agg[N*HID]
    char* ws = (char*)d_ws;
    size_t off = 0;
    auto carve = [&](size_t bytes) -> float* {
        float* p = (float*)(ws + off);
        off = (off + bytes + 255) & ~(size_t)255;
        return p;
    };
    float* deg  = carve((size_t)N * 4);
    float* dinv = carve((size_t)N * 4);
    float* h    = carve((size_t)N * HID * 4);
    float* agg  = carve((size_t)N * HID * 4);
    float* g    = agg;   // layer-2 GEMM output reuses agg buffer

    const int TPB = 256;
    auto cdiv = [](long long a, long long b) { return (int)((a + b - 1) / b); };

    // 1) symmetric-norm coefficients
    k_deg_init <<<cdiv(N, TPB), TPB, 0, stream>>>(deg, N);
    k_deg_count<<<cdiv(E, TPB), TPB, 0, stream>>>(dstp, deg, E);
    k_dinv     <<<cdiv(N, TPB), TPB, 0, stream>>>(deg, dinv, N);

    // 2) layer 1 GEMM: h = x @ W1   (N x IN) @ (IN x HID)
    {
        long long tiles = (long long)(N / 16) * (HID / 16);
        k_gemm_wmma_f32<<<cdiv(tiles * 32, TPB), TPB, 0, stream>>>(
            x, W1, h, N, HID, IN);
    }

    // 3) layer 1 aggregation: agg = D^-1/2 (A+I) D^-1/2 h + b1, then ReLU -> h
    k_init_agg<<<cdiv((long long)N * HID, TPB), TPB, 0, stream>>>(
        h, dinv, b1, agg, N, HID);
    k_edge_scatter<<<cdiv((long long)E * 32, TPB), TPB, 0, stream>>>(
        srcp, dstp, h, dinv, agg, E, HID);
    k_relu<<<cdiv((long long)N * HID, TPB), TPB, 0, stream>>>(
        agg, h, N * HID);

    // 4) layer 2 GEMM: g = h @ W2   (N x HID) @ (HID x OUT)
    {
        long long tiles = (long long)(N / 16) * (OUT / 16);
        k_gemm_wmma_f32<<<cdiv(tiles * 32, TPB), TPB, 0, stream>>>(
            h, W2, g, N, OUT, HID);
    }

    // 5) layer 2 aggregation directly into d_out (bias folded into init)
    k_init_agg<<<cdiv((long long)N * OUT, TPB), TPB, 0, stream>>>(
        g, dinv, b2, out, N, OUT);
    k_edge_scatter<<<cdiv((long long)E * 32, TPB), TPB, 0, stream>>>(
        srcp, dstp, g, dinv, out, E, OUT);
}